// ClusterMemory_40956808134724
// MI455X (gfx1250) — compile-verified
//
#include <hip/hip_runtime.h>

// ClusterMemory forward + EMA bank update for MI455X (gfx1250, wave32).
// B=2048, D=256, N=65536.  f32 end-to-end using V_WMMA_F32_16X16X4_F32.

typedef __attribute__((ext_vector_type(2))) float v2f;
typedef __attribute__((ext_vector_type(8))) float v8f;

#define B_SZ 2048
#define D_SZ 256
#define N_SZ 65536
#define INV_TEMP 20.0f   // 1 / 0.05
#define MOM 0.2f
#define NORM_EPS 1e-12f

// ---------------------------------------------------------------------------
// Kernel 1: row-normalize inputs -> xn (one wave per row, 8 elems/lane)
// ---------------------------------------------------------------------------
__global__ __launch_bounds__(32) void normalize_rows_kernel(
    const float* __restrict__ x, float* __restrict__ xn) {
  const int row = blockIdx.x;
  const int lane = threadIdx.x;
  const float* p = x + (size_t)row * D_SZ;
  float v[8];
  float s = 0.f;
#pragma unroll
  for (int t = 0; t < 8; ++t) {
    v[t] = p[lane + 32 * t];
    s += v[t] * v[t];
  }
#pragma unroll
  for (int off = 16; off > 0; off >>= 1) s += __shfl_xor(s, off, 32);
  const float inv = 1.0f / fmaxf(sqrtf(s), NORM_EPS);
  float* q = xn + (size_t)row * D_SZ;
#pragma unroll
  for (int t = 0; t < 8; ++t) q[lane + 32 * t] = v[t] * inv;
}

// ---------------------------------------------------------------------------
// Kernel 2: outputs[B,N] = (xn @ features^T) * 20 via V_WMMA_F32_16X16X4_F32.
// Each wave owns a 32(M) x 64(N) tile: 2 A-subtiles x 4 B-subtiles,
// 8 v8f accumulators.  Per K-step: 2 A loads + 4 B loads feed 8 WMMAs
// (load:wmma = 0.75).  Sub-tile rows reached via 16*D*4=16384-byte
// immediate offsets so only two base pointers advance.
//
// A 16x4 frag : lane holds row m0+(lane&15), k = k0 + 2*(lane>>4) + {0,1}
// B 4x16 frag : lane holds col n0+(lane&15), k = k0 + 2*(lane>>4) + {0,1}
//               (B = features^T, so that's features row n, contiguous in k)
// C  16x16    : VGPR v <-> row m0 + v + 8*(lane>>4), col n0 + (lane&15)
// ---------------------------------------------------------------------------
__global__ __launch_bounds__(256) void cosine_gemm_wmma_kernel(
    const float* __restrict__ xn, const float* __restrict__ feat,
    float* __restrict__ out) {
  const int lane = threadIdx.x & 31;
  const int wid = blockIdx.x * 8 + (threadIdx.x >> 5);
  const int m0 = (wid & 63) * 32;   // 64 M-tiles (inner: reuse B strip in L2)
  const int n0 = (wid >> 6) * 64;   // 1024 N-strips of width 64

  const int row_sel = lane & 15;
  const int k_sel = (lane >> 4) << 1;

  const float* ap = xn + (size_t)(m0 + row_sel) * D_SZ + k_sel;
  const float* bp = feat + (size_t)(n0 + row_sel) * D_SZ + k_sel;

  v8f c00 = {}, c01 = {}, c02 = {}, c03 = {};
  v8f c10 = {}, c11 = {}, c12 = {}, c13 = {};
#pragma unroll 4
  for (int k = 0; k < D_SZ; k += 4) {
    const v2f a0 = *(const v2f*)(ap + k);
    const v2f a1 = *(const v2f*)(ap + 16 * D_SZ + k);
    const v2f b0 = *(const v2f*)(bp + k);
    const v2f b1 = *(const v2f*)(bp + 16 * D_SZ + k);
    const v2f b2 = *(const v2f*)(bp + 32 * D_SZ + k);
    const v2f b3 = *(const v2f*)(bp + 48 * D_SZ + k);
    c00 = __builtin_amdgcn_wmma_f32_16x16x4_f32(false, a0, false, b0, (short)0, c00, false, false);
    c01 = __builtin_amdgcn_wmma_f32_16x16x4_f32(false, a0, false, b1, (short)0, c01, false, false);
    c02 = __builtin_amdgcn_wmma_f32_16x16x4_f32(false, a0, false, b2, (short)0, c02, false, false);
    c03 = __builtin_amdgcn_wmma_f32_16x16x4_f32(false, a0, false, b3, (short)0, c03, false, false);
    c10 = __builtin_amdgcn_wmma_f32_16x16x4_f32(false, a1, false, b0, (short)0, c10, false, false);
    c11 = __builtin_amdgcn_wmma_f32_16x16x4_f32(false, a1, false, b1, (short)0, c11, false, false);
    c12 = __builtin_amdgcn_wmma_f32_16x16x4_f32(false, a1, false, b2, (short)0, c12, false, false);
    c13 = __builtin_amdgcn_wmma_f32_16x16x4_f32(false, a1, false, b3, (short)0, c13, false, false);
  }

  const int voff = (lane >> 4) << 3;     // 0 or 8
  const int col = lane & 15;
  v8f acc[2][4] = {{c00, c01, c02, c03}, {c10, c11, c12, c13}};
#pragma unroll
  for (int i = 0; i < 2; ++i) {
#pragma unroll
    for (int j = 0; j < 4; ++j) {
      float* o = out + (size_t)(m0 + 16 * i + voff) * N_SZ + n0 + 16 * j + col;
#pragma unroll
      for (int v = 0; v < 8; ++v) {
        o[(size_t)v * N_SZ] = acc[i][j][v] * INV_TEMP;
      }
    }
  }
}

// ---------------------------------------------------------------------------
// Kernel 3: sequential momentum-EMA scatter, chain-parallel form.
// One wave per batch index i; only the first occurrence of targets[i]
// proceeds and processes every later occurrence of that target in order,
// exactly matching the lax.scan write-after-write semantics.
// new_features must already contain a copy of features (d2d memcpy).
// ---------------------------------------------------------------------------
__global__ __launch_bounds__(32) void ema_update_kernel(
    const float* __restrict__ xn, const int* __restrict__ targets,
    const float* __restrict__ features, float* __restrict__ new_features) {
  const int i = blockIdx.x;
  const int y = targets[i];
  // first-occurrence test (uniform scalar loop, cheap: <= B iterations)
  for (int j = 0; j < i; ++j)
    if (targets[j] == y) return;

  const int lane = threadIdx.x;
  float f[8];
  const float* fp = features + (size_t)y * D_SZ;
#pragma unroll
  for (int t = 0; t < 8; ++t) f[t] = fp[lane + 32 * t];

  for (int j = i; j < B_SZ; ++j) {
    if (targets[j] != y) continue;
    const float* xp = xn + (size_t)j * D_SZ;
    float s = 0.f;
#pragma unroll
    for (int t = 0; t < 8; ++t) {
      f[t] = MOM * f[t] + (1.0f - MOM) * xp[lane + 32 * t];
      s += f[t] * f[t];
    }
#pragma unroll
    for (int off = 16; off > 0; off >>= 1) s += __shfl_xor(s, off, 32);
    const float inv = 1.0f / sqrtf(s);   // reference renorm has no eps
#pragma unroll
    for (int t = 0; t < 8; ++t) f[t] *= inv;
  }

  float* op = new_features + (size_t)y * D_SZ;
#pragma unroll
  for (int t = 0; t < 8; ++t) op[lane + 32 * t] = f[t];
}

// ---------------------------------------------------------------------------
extern "C" void kernel_launch(void* const* d_in, const int* in_sizes, int n_in,
                              void* d_out, int out_size, void* d_ws, size_t ws_size,
                              hipStream_t stream) {
  const float* inputs   = (const float*)d_in[0];   // [B, D]
  const int*   targets  = (const int*)d_in[1];     // [B]   (integer -> int*)
  const float* features = (const float*)d_in[2];   // [N, D]

  float* out_logits = (float*)d_out;                          // [B, N]
  float* out_feats  = out_logits + (size_t)B_SZ * N_SZ;       // [N, D]
  float* xn         = (float*)d_ws;                           // [B, D] scratch

  // 1) xn = normalize(inputs)
  normalize_rows_kernel<<<B_SZ, 32, 0, stream>>>(inputs, xn);

  // 2) logits = xn @ features^T / TEMP   (WMMA f32 16x16x4, 32x64 per wave)
  //    64 M-tiles * 1024 N-strips = 65536 waves / 8 per block
  cosine_gemm_wmma_kernel<<<8192, 256, 0, stream>>>(xn, features, out_logits);

  // 3) new_features = features, then apply EMA chains in place
  hipMemcpyAsync(out_feats, features, (size_t)N_SZ * D_SZ * sizeof(float),
                 hipMemcpyDeviceToDevice, stream);
  ema_update_kernel<<<B_SZ, 32, 0, stream>>>(xn, targets, features, out_feats);
}